// LossLayer_9809705304582
// MI455X (gfx1250) — compile-verified
//
#include <hip/hip_runtime.h>
#include <math.h>

// Problem constants from the reference (not checked against in_sizes at launch).
#define BATCH      8
#define M_MAX      64      // annotations per image
#define NTHREADS   256     // 8 wave32s per block

typedef __attribute__((ext_vector_type(2))) float v2f;
typedef __attribute__((ext_vector_type(8))) float v8f;

// ---------------------------------------------------------------------------
// Wave32 sum via V_WMMA_F32_16X16X4_F32.
// A (16x4 f32): lanes 0-15 hold K0,K1 (VGPR0,1); lanes 16-31 hold K2,K3.
// We place lane's value in element0, 0 in element1; B = all ones.
// Then D[m][n] = x[m] + x[m+16] for every n. Lane n<16 holds rows 0..7 of
// column n, lane n>=16 holds rows 8..15 => sum of own 8 regs + shfl_xor(16)
// = full 32-lane sum, in every lane.
// ---------------------------------------------------------------------------
__device__ __forceinline__ float wave_sum_wmma(float x) {
  v2f a; a[0] = x;     a[1] = 0.0f;
  v2f b; b[0] = 1.0f;  b[1] = 1.0f;
  v8f c = {};
  c = __builtin_amdgcn_wmma_f32_16x16x4_f32(false, a, false, b,
                                            (short)0, c, false, false);
  float s = c[0] + c[1] + c[2] + c[3] + c[4] + c[5] + c[6] + c[7];
  s += __shfl_xor(s, 16, 32);
  return s;
}

// Block-level sum (deterministic, fixed order). lds must hold >= 8 floats.
// Must be called by all threads of the block (uniform control flow).
__device__ __forceinline__ float block_sum(float v, float* lds) {
  float w = wave_sum_wmma(v);
  const int lane = threadIdx.x & 31;
  const int wid  = threadIdx.x >> 5;
  if (lane == 0) lds[wid] = w;
  __syncthreads();
  if (threadIdx.x == 0) {
    float r = 0.0f;
    for (int i = 0; i < NTHREADS / 32; ++i) r += lds[i];
    lds[0] = r;
  }
  __syncthreads();
  float r = lds[0];
  __syncthreads();
  return r;
}

__device__ __forceinline__ float sl1(float x) {  // smooth L1
  float ax = fabsf(x);
  return (ax < 1.0f) ? 0.5f * ax * ax : ax - 0.5f;
}

// Monotonic float<->uint mapping (order-preserving for all finite + +-inf).
__device__ __forceinline__ unsigned f2mono(float f) {
  unsigned u = __float_as_uint(f);
  return (u & 0x80000000u) ? ~u : (u | 0x80000000u);
}
__device__ __forceinline__ float mono2f(unsigned m) {
  unsigned u = (m & 0x80000000u) ? (m & 0x7FFFFFFFu) : ~m;
  return __uint_as_float(u);
}

// ---------------------------------------------------------------------------
// Stage 1: per (image, anchor-chunk) block.
//  - annotations staged in LDS (64x14 f32 = 3.5 KB, broadcast reads)
//  - per-anchor IoU max/argmax over 64 annotations
//  - pos/neg stats, smooth-L1 box & landmark partial sums
//  - per-anchor monotonic key for hard-negative top-k  (neg ? -cls1 : -inf)
//  - bbox/ldm regressions read ONLY for pos anchors (saves ~30 MB traffic)
//  - 6 accumulators block-reduced via WMMA, written as per-block partials
// ---------------------------------------------------------------------------
__global__ void retina_loss_stage1(const float* __restrict__ cls,
                                   const float* __restrict__ breg,
                                   const float* __restrict__ lreg,
                                   const float* __restrict__ anch,
                                   const float* __restrict__ ann,
                                   unsigned* __restrict__ keys,
                                   float* __restrict__ partials,
                                   int A, int M, int nblk) {
  __shared__ float s_ann[M_MAX * 14];
  __shared__ float s_areaB[M_MAX];
  __shared__ float s_red[NTHREADS / 32];

  const int b = blockIdx.y;
  const int a = blockIdx.x * NTHREADS + threadIdx.x;

  for (int i = threadIdx.x; i < M * 14; i += NTHREADS)
    s_ann[i] = ann[(size_t)b * M * 14 + i];
  __syncthreads();
  if (threadIdx.x < M) {
    const float* r = &s_ann[threadIdx.x * 14];
    s_areaB[threadIdx.x] = (r[2] - r[0]) * (r[3] - r[1]);
  }
  __syncthreads();

  float posc = 0.0f, poss = 0.0f, negc = 0.0f;
  float bls = 0.0f, lls = 0.0f, ldmc = 0.0f;

  if (a < A) {
    const float4 an = ((const float4*)anch)[a];
    const float aw = an.z - an.x, ah = an.w - an.y;
    const float acx = an.x + 0.5f * aw, acy = an.y + 0.5f * ah;
    const float areaA = aw * ah;

    // iou_max / argmax (first-max tie-break, matching jnp.argmax)
    float best = -__builtin_inff();
    int barg = 0;
    for (int m = 0; m < M; ++m) {
      const float* r = &s_ann[m * 14];
      float v;
      if (r[0] > 0.0f) {
        float iw = fminf(an.z, r[2]) - fmaxf(an.x, r[0]); iw = fmaxf(iw, 0.0f);
        float ih = fminf(an.w, r[3]) - fmaxf(an.y, r[1]); ih = fmaxf(ih, 0.0f);
        float inter = iw * ih;
        float ua = fmaxf(areaA + s_areaB[m] - inter, 1e-8f);
        v = inter / ua;
      } else {
        v = -1.0f;
      }
      if (v > best) { best = v; barg = m; }
    }

    const bool pos = best >= 0.5f;
    const bool neg = best < 0.3f;

    const float2 c2 = ((const float2*)cls)[(size_t)b * A + a];
    if (pos) { posc = 1.0f; poss = -c2.x; }
    if (neg) negc = 1.0f;
    keys[(size_t)b * A + a] = f2mono(neg ? -c2.y : -__builtin_inff());

    if (pos) {
      const float* r = &s_ann[barg * 14];
      const float gw = r[2] - r[0], gh = r[3] - r[1];
      const float gcx = r[0] + 0.5f * gw, gcy = r[1] + 0.5f * gh;
      const float dx = (gcx - acx) / (aw + 1e-14f);
      const float dy = (gcy - acy) / (ah + 1e-14f);
      const float dw = logf(gw / aw);
      const float dh = logf(gh / ah);
      const float4 br = ((const float4*)breg)[(size_t)b * A + a];
      bls = sl1(dx / 0.1f - br.x) + sl1(dy / 0.1f - br.y) +
            sl1(dw / 0.2f - br.z) + sl1(dh / 0.2f - br.w);

      float lsum = 0.0f;
      for (int j = 0; j < 10; ++j) lsum += r[4 + j];
      if (lsum > 0.0f) {
        ldmc = 1.0f;
        const float* lp = &lreg[((size_t)b * A + a) * 10];
        for (int j = 0; j < 5; ++j) {
          float ltx = (r[4 + 2 * j]     - acx) / (aw + 1e-14f) / 0.1f;
          float lty = (r[4 + 2 * j + 1] - acy) / (ah + 1e-14f) / 0.1f;
          lls += sl1(ltx - lp[2 * j]) + sl1(lty - lp[2 * j + 1]);
        }
      }
    }
  }

  // Block reduction (all threads participate; WMMA path).
  float acc6[6] = {posc, poss, negc, bls, lls, ldmc};
  for (int j = 0; j < 6; ++j) {
    float s = block_sum(acc6[j], s_red);
    if (threadIdx.x == 0)
      partials[((size_t)b * nblk + blockIdx.x) * 6 + j] = s;
  }
}

// ---------------------------------------------------------------------------
// Stage 2: one block per image.
//  - reduce per-block partials (fixed order)
//  - exact top-k sum of neg losses by 4-pass radix select (8-bit digits)
//    over monotonic keys, tie-aware
//  - final loss assembly
// ---------------------------------------------------------------------------
__global__ void retina_loss_stage2(const unsigned* __restrict__ keys,
                                   const float* __restrict__ partials,
                                   float* __restrict__ out,
                                   int A, int nblk) {
  const int b = blockIdx.x;
  __shared__ float    s_red[NTHREADS / 32];
  __shared__ unsigned s_hist[256];
  __shared__ unsigned s_sel[2];
  __shared__ float    S[6];

  for (int j = 0; j < 6; ++j) {
    float l = 0.0f;
    for (int i = threadIdx.x; i < nblk; i += NTHREADS)
      l += partials[((size_t)b * nblk + i) * 6 + j];
    float s = block_sum(l, s_red);
    if (threadIdx.x == 0) S[j] = s;
    __syncthreads();
  }

  const int np   = (int)(S[0] + 0.5f);
  const float poss = S[1];
  const int negc = (int)(S[2] + 0.5f);
  const float bls = S[3], lls = S[4];
  const int nldm = (int)(S[5] + 0.5f);
  int k = 3 * np; if (negc < k) k = negc;

  const unsigned* kb = keys + (size_t)b * A;
  float neg_mean = 0.0f;

  if (np > 0 && k > 0) {
    unsigned prefix = 0u;
    unsigned remaining = (unsigned)k;
    for (int shift = 24; shift >= 0; shift -= 8) {
      s_hist[threadIdx.x] = 0u;  // NTHREADS == 256
      __syncthreads();
      const unsigned hm = (shift == 24) ? 0u : (0xFFFFFFFFu << (shift + 8));
      for (int i = threadIdx.x; i < A; i += NTHREADS) {
        unsigned key = kb[i];
        if ((key & hm) == (prefix & hm))
          atomicAdd(&s_hist[(key >> shift) & 0xFFu], 1u);
      }
      __syncthreads();
      if (threadIdx.x == 0) {
        unsigned cum = 0; unsigned chosen = 0; unsigned rem2 = remaining;
        for (int bin = 255; bin >= 0; --bin) {
          unsigned h = s_hist[bin];
          if (cum + h >= remaining) { chosen = (unsigned)bin; rem2 = remaining - cum; break; }
          cum += h;
        }
        s_sel[0] = chosen; s_sel[1] = rem2;
      }
      __syncthreads();
      prefix |= (s_sel[0] << shift);
      remaining = s_sel[1];
      __syncthreads();
    }
    const unsigned T = prefix;  // k-th largest key (threshold)
    float loc = 0.0f;
    for (int i = threadIdx.x; i < A; i += NTHREADS) {
      unsigned key = kb[i];
      if (key > T) loc += mono2f(key);
    }
    const float above = block_sum(loc, s_red);
    const float neg_sum = above + (float)remaining * mono2f(T);
    neg_mean = neg_sum / (float)(k > 1 ? k : 1);
  }

  if (threadIdx.x == 0) {
    float cls_loss = 0.0f, bl = 0.0f, ll = 0.0f;
    if (np > 0) {
      cls_loss = poss / (float)(np > 1 ? np : 1) + neg_mean;
      int d = np * 4; bl = bls / (float)(d > 1 ? d : 1);
    }
    if (nldm > 0) {
      int d = nldm * 10; ll = lls / (float)(d > 1 ? d : 1);
    }
    out[b]             = cls_loss;
    out[BATCH + b]     = bl;
    out[2 * BATCH + b] = ll;
  }
}

// ---------------------------------------------------------------------------
extern "C" void kernel_launch(void* const* d_in, const int* in_sizes, int n_in,
                              void* d_out, int out_size, void* d_ws, size_t ws_size,
                              hipStream_t stream) {
  (void)n_in; (void)out_size; (void)ws_size;
  const float* cls  = (const float*)d_in[0];  // (B, A, 2)
  const float* breg = (const float*)d_in[1];  // (B, A, 4)
  const float* lreg = (const float*)d_in[2];  // (B, A, 10)
  const float* anch = (const float*)d_in[3];  // (1, A, 4)
  const float* ann  = (const float*)d_in[4];  // (B, M, 14)

  const int A = in_sizes[3] / 4;                        // 67200
  const int M = in_sizes[4] / (BATCH * 14);             // 64
  const int nblk = (A + NTHREADS - 1) / NTHREADS;       // 263

  unsigned* keys   = (unsigned*)d_ws;                       // B*A u32
  float* partials  = (float*)d_ws + (size_t)BATCH * A;      // B*nblk*6 f32

  dim3 g1(nblk, BATCH);
  retina_loss_stage1<<<g1, NTHREADS, 0, stream>>>(cls, breg, lreg, anch, ann,
                                                  keys, partials, A, M, nblk);
  retina_loss_stage2<<<BATCH, NTHREADS, 0, stream>>>(keys, partials,
                                                     (float*)d_out, A, nblk);
}